// NTfm3DDelta_73074573574187
// MI455X (gfx1250) — compile-verified
//
#include <hip/hip_runtime.h>

typedef __attribute__((ext_vector_type(2))) float v2f;
typedef __attribute__((ext_vector_type(8))) float v8f;

#define BATCH 16
#define KMASK 8
#define HW    (240 * 320)                    // 76800 pixels per batch image
#define CHUNKS 150                           // pixel chunks per batch
#define PIX_PER_CHUNK (HW / CHUNKS)          // 512
#define WAVES 8
#define ITERS (PIX_PER_CHUNK / (WAVES * 16)) // 4

__global__ __launch_bounds__(256)
void NTfm3DDelta_kernel(const float* __restrict__ points,
                        const float* __restrict__ masks,
                        const float* __restrict__ transforms,
                        float* __restrict__ out)
{
    const int tid  = threadIdx.x;
    const int lane = tid & 31;
    const int wave = tid >> 5;
    const int b    = blockIdx.y;     // batch
    const int chunk = blockIdx.x;    // pixel chunk within batch

    const int m  = lane & 15;        // A: row i' (0..15) | B/C/D: pixel column
    const int hi = lane >> 4;        // half-wave selector

    // ---- Build A-matrix registers: A[i',k] = (R_k - I | t_k), rows 12..15 = 0.
    // 16x4 f32 A layout: lanes 0-15 -> K0(K4) in v.x, K1(K5) in v.y;
    //                    lanes 16-31 -> K2(K6), K3(K7).
    const int kv0 = hi * 2;
    const int kv1 = hi * 2 + 1;
    // transforms[b,k,:,:] is 12 contiguous floats; i' indexes it directly.
    const float diag = (m == 0 || m == 5 || m == 10) ? 1.0f : 0.0f;
    const float* T = transforms + b * (KMASK * 12);
    const bool valid = (m < 12);
    v2f a_lo, a_hi2;
    a_lo.x  = valid ? (T[ kv0      * 12 + m] - diag) : 0.0f;
    a_lo.y  = valid ? (T[ kv1      * 12 + m] - diag) : 0.0f;
    a_hi2.x = valid ? (T[(kv0 + 4) * 12 + m] - diag) : 0.0f;
    a_hi2.y = valid ? (T[(kv1 + 4) * 12 + m] - diag) : 0.0f;

    // Uniform base pointers + 32-bit per-lane offsets -> SADDR addressing.
    const float* Mb = masks  + b * (KMASK * HW);
    const float* Pb = points + b * (3 * HW);
    float*       Ob = out    + b * (3 * HW);

    // Loop-invariant plane offsets (all < 2^20, comfortably int).
    const int om0 = kv0 * HW;            // mask plane for B vgpr0, wmma #1
    const int om1 = kv1 * HW;            // mask plane for B vgpr1, wmma #1
    const int om2 = (kv0 + 4) * HW;      // wmma #2
    const int om3 = (kv1 + 4) * HW;
    const int orow = hi ? 2 * HW : 0;    // output row for the uniform store

    int pix = chunk * PIX_PER_CHUNK + wave * 16 + m;
    #pragma unroll
    for (int it = 0; it < ITERS; ++it, pix += WAVES * 16) {
        // ---- B-matrix (masks): 4x16 per WMMA, column = pixel, row = k plane.
        v2f b_lo, b_hi2;
        b_lo.x  = Mb[om0 + pix];
        b_lo.y  = Mb[om1 + pix];
        b_hi2.x = Mb[om2 + pix];
        b_hi2.y = Mb[om3 + pix];

        // ---- Blend transforms with masks on the matrix pipe:
        // M[i',pix] = sum_{k=0..7} A[i',k] * mask[k,pix]
        v8f c = {};
        c = __builtin_amdgcn_wmma_f32_16x16x4_f32(false, a_lo,  false, b_lo,
                                                  (short)0, c, false, false);
        c = __builtin_amdgcn_wmma_f32_16x16x4_f32(false, a_hi2, false, b_hi2,
                                                  (short)0, c, false, false);

        // ---- Apply blended 3x4 delta per pixel.
        const float p0 = Pb[pix];
        const float p1 = Pb[HW + pix];
        const float p2 = Pb[2 * HW + pix];

        // lanes 0-15: c[0..3] = row i=0 coeffs, c[4..7] = row i=1
        // lanes 16-31: c[0..3] = row i=2 coeffs
        const float pa   = hi ? p2 : p0;
        const float outa = pa + c[0] * p0 + c[1] * p1 + c[2] * p2 + c[3];
        const float outb = p1 + c[4] * p0 + c[5] * p1 + c[6] * p2 + c[7];

        Ob[orow + pix] = outa;                 // row 0 (lo lanes) / row 2 (hi lanes)
        if (!hi) Ob[HW + pix] = outb;          // row 1 (lo lanes only)
    }
}

extern "C" void kernel_launch(void* const* d_in, const int* in_sizes, int n_in,
                              void* d_out, int out_size, void* d_ws, size_t ws_size,
                              hipStream_t stream) {
    const float* points     = (const float*)d_in[0]; // (16,3,240,320)
    const float* masks      = (const float*)d_in[1]; // (16,8,240,320)
    const float* transforms = (const float*)d_in[2]; // (16,8,3,4)
    float* out = (float*)d_out;                      // (16,3,240,320)

    dim3 grid(CHUNKS, BATCH);
    dim3 block(256);
    NTfm3DDelta_kernel<<<grid, block, 0, stream>>>(points, masks, transforms, out);
}